// ForwardDiffusion_90374701842590
// MI455X (gfx1250) — compile-verified
//
#include <hip/hip_runtime.h>

typedef __attribute__((ext_vector_type(2))) float v2f;
typedef __attribute__((ext_vector_type(4))) float v4f;
typedef __attribute__((ext_vector_type(8))) float v8f;

#define FD_STEPS   1000
#define FD_LENGTH  1024
#define FD_BATCH   64
#define FD_DT      0.001f
#define FD_A       (1.0f - FD_DT)            // OU decay per step, a = 0.999
#define FD_C       0.015811388300841896f     // SIGMA0 * sqrt(dt)
#define FD_NSEG    63                        // ceil(999 / 16) segments of 16 steps

// ---------------------------------------------------------------------------
// One 16-step scan segment computed with the matrix unit:
//   S_{s0+1+i} = sum_{j<=i} (c * a^{i-j}) * z_{s0+j}  +  a^{i+1} * S_{s0}
// = L'(16x16, lower-triangular, constant, with c folded in) x Z(16x16 raw
// noise tile) + carry, evaluated as 4 chained V_WMMA_F32_16X16X4_F32 with the
// carry pre-loaded into the accumulator. Folding c into A lets the noise
// loads feed the WMMAs directly (no per-segment scaling VALU).
// TAIL=true adds range guards (needed only for the final segment).
// ---------------------------------------------------------------------------
template <bool TAIL>
__device__ __forceinline__ void fd_segment(
    const float* __restrict__ noise, float* __restrict__ S,
    const v2f aop[4], const float capw[8], float& Sprev,
    const int s0, const int half, const int n, const int col) {
  // B operand (4x16 f32 per wmma): row j of B = noise[s0 + j][col];
  // lanes 0-15 hold rows {4k,4k+1}, lanes 16-31 rows {4k+2,4k+3}, N = n.
  v2f bop[4];
#pragma unroll
  for (int k = 0; k < 4; ++k) {
    const int j0 = 4 * k + 2 * half;
    if (TAIL) {
      bop[k].x = (s0 + j0     < FD_STEPS - 1)
          ? noise[(size_t)(s0 + j0)     * FD_LENGTH + col] : 0.0f;
      bop[k].y = (s0 + j0 + 1 < FD_STEPS - 1)
          ? noise[(size_t)(s0 + j0 + 1) * FD_LENGTH + col] : 0.0f;
    } else {
      bop[k].x = noise[(size_t)(s0 + j0)     * FD_LENGTH + col];
      bop[k].y = noise[(size_t)(s0 + j0 + 1) * FD_LENGTH + col];
    }
  }

  // Accumulator init: carry term a^(M+1) * S_{s0}, M = v + 8*half.
  v8f c;
#pragma unroll
  for (int v = 0; v < 8; ++v) c[v] = capw[v] * Sprev;

  // D = L' * Z + C over K = 16 (4 chained K=4 f32 WMMAs).
#pragma unroll
  for (int k = 0; k < 4; ++k)
    c = __builtin_amdgcn_wmma_f32_16x16x4_f32(
            /*neg_a=*/false, aop[k], /*neg_b=*/false, bop[k],
            /*c_mod=*/(short)0, c, /*reuse_a=*/false, /*reuse_b=*/false);

  // D row M = v + 8*half, col N = n  ->  S[s0 + 1 + M][col]
#pragma unroll
  for (int v = 0; v < 8; ++v) {
    const int s = s0 + 1 + v + 8 * half;
    if (!TAIL || s < FD_STEPS) S[(size_t)s * FD_LENGTH + col] = c[v];
  }

  // Carry: row 15 of D lives in VGPR 7 of lanes 16..31; broadcast to both
  // halves' column-n lanes.
  Sprev = __shfl(c[7], 16 + n, 32);
}

// ---------------------------------------------------------------------------
// Kernel 1: batch-independent noise accumulation S[s][l], s = 1..999, via the
// WMMA segmented linear-recurrence scan. One wave per 16-column tile (64
// tiles). Chain length per column: 63 WMMA segments instead of 999 scalar
// FMAs. Segments 0..61 are branch-free; only the last segment is guarded.
// ---------------------------------------------------------------------------
__global__ __launch_bounds__(32)
void fd_wmma_scan(const float* __restrict__ noise, float* __restrict__ S) {
  const int lane = threadIdx.x;
  const int half = lane >> 4;          // 0: lanes 0-15, 1: lanes 16-31
  const int n    = lane & 15;
  const int col  = blockIdx.x * 16 + n;

  // A operand (16x4 f32 per wmma): lanes 0-15 hold K={4k,4k+1}, lanes 16-31
  // hold K={4k+2,4k+3}; M = n for both halves.
  // L'[i][j] = (j<=i) ? c * a^(i-j) : 0   (noise scale c folded in).
  v2f aop[4];
#pragma unroll
  for (int k = 0; k < 4; ++k) {
    const int j0 = 4 * k + 2 * half;
    aop[k].x = (j0     <= n) ? FD_C * __powf(FD_A, (float)(n - j0))     : 0.0f;
    aop[k].y = (j0 + 1 <= n) ? FD_C * __powf(FD_A, (float)(n - j0 - 1)) : 0.0f;
  }

  // Carry scale for accumulator init: C row M = v + 8*half needs a^(M+1).
  float capw[8];
#pragma unroll
  for (int v = 0; v < 8; ++v)
    capw[v] = __powf(FD_A, (float)(v + 1 + 8 * half));

  float Sprev = 0.0f;  // S at the segment start (per column n)

  // Steady state: fully in-range (max step 992 <= 999, max noise row 991).
  for (int seg = 0; seg < FD_NSEG - 1; ++seg)
    fd_segment<false>(noise, S, aop, capw, Sprev, seg * 16, half, n, col);

  // Tail segment (steps 993..1008 clipped to 999, noise rows clipped to 998).
  fd_segment<true>(noise, S, aop, capw, Sprev, (FD_NSEG - 1) * 16, half, n, col);
}

// ---------------------------------------------------------------------------
// Kernel 2: out[b][s][l] = a^s * x[b][l] + S[s][l]  (s=0 row is x itself).
// 256 MB streaming output -> non-temporal b128 stores so the write-once data
// does not evict the L2-resident S (4 MB) and x (256 KB). This kernel is the
// roofline-limiting phase: ~11 us at 23.3 TB/s.
// ---------------------------------------------------------------------------
__global__ __launch_bounds__(256)
void fd_broadcast(const float* __restrict__ x, const float* __restrict__ S,
                  float* __restrict__ out) {
  const int s = blockIdx.x;
  const int b = blockIdx.y;
  const int l = threadIdx.x << 2;

  const v4f xv = *reinterpret_cast<const v4f*>(x + (size_t)b * FD_LENGTH + l);
  v4f o;
  if (s == 0) {
    o = xv;
  } else {
    const float alpha = __powf(FD_A, (float)s);
    const v4f sv = *reinterpret_cast<const v4f*>(S + (size_t)s * FD_LENGTH + l);
    o.x = fmaf(alpha, xv.x, sv.x);
    o.y = fmaf(alpha, xv.y, sv.y);
    o.z = fmaf(alpha, xv.z, sv.z);
    o.w = fmaf(alpha, xv.w, sv.w);
  }
  v4f* dst = reinterpret_cast<v4f*>(out + ((size_t)b * FD_STEPS + s) * FD_LENGTH + l);
  __builtin_nontemporal_store(o, dst);
}

// ---------------------------------------------------------------------------
extern "C" void kernel_launch(void* const* d_in, const int* in_sizes, int n_in,
                              void* d_out, int out_size, void* d_ws, size_t ws_size,
                              hipStream_t stream) {
  const float* x     = (const float*)d_in[0];   // [64, 1024] f32
  const float* noise = (const float*)d_in[1];   // [999, 1024] f32
  float* out = (float*)d_out;                   // [64, 1000, 1024] f32
  float* S   = (float*)d_ws;                    // scratch: 1000*1024*4 = 4.1 MB

  // Phase 1: batch-independent scan (writes S rows 1..999; row 0 unused).
  fd_wmma_scan<<<dim3(FD_LENGTH / 16), dim3(32), 0, stream>>>(noise, S);

  // Phase 2: bandwidth-bound broadcast into the 256 MB trajectory.
  fd_broadcast<<<dim3(FD_STEPS, FD_BATCH), dim3(FD_LENGTH / 4), 0, stream>>>(x, S, out);
}